// InverseLoss_48266842472944
// MI455X (gfx1250) — compile-verified
//
#include <hip/hip_runtime.h>
#include <hip/hip_bf16.h>

typedef __attribute__((ext_vector_type(2))) float v2f;
typedef __attribute__((ext_vector_type(8))) float v8f;

#define H_  256
#define W_  256
#define WP_ 257   // (W+1), row stride of displacement grid
#define RH_ 254
#define RW_ 254
#define NB_ 128   // blocks in main kernel (128*256 = 32768 threads -> exactly 2 iters)

// ---------------------------------------------------------------------------
// Wave-wide sum via V_WMMA_F32_16X16X4_F32.
// A = all-ones 16x4, C = 0  =>  D[m,n] = sum_k B[k,n] for every row m.
// Sum over all columns == sum over ALL elements of B, which is invariant to
// whatever lane/VGPR striping the HW uses for B, so placing each lane's
// partial as {x, 0} is correct regardless of the exact B layout.
// Every 16-lane half of D's first VGPR contains all 16 column sums, so a
// 4-step xor-shuffle yields the full wave sum in every lane.
// ---------------------------------------------------------------------------
__device__ __forceinline__ float wave_reduce_sum_wmma(float x) {
  v2f a; a[0] = 1.0f; a[1] = 1.0f;     // A matrix: all ones (layout-proof)
  v2f b; b[0] = x;    b[1] = 0.0f;     // each lane contributes its partial once
  v8f c = {};
  v8f d = __builtin_amdgcn_wmma_f32_16x16x4_f32(
      /*neg_a=*/false, a, /*neg_b=*/false, b,
      /*c_mod=*/(short)0, c, /*reuse_a=*/false, /*reuse_b=*/false);
  float s = d[0];
  s += __shfl_xor(s, 1, 32);
  s += __shfl_xor(s, 2, 32);
  s += __shfl_xor(s, 4, 32);
  s += __shfl_xor(s, 8, 32);
  return s;  // wave total, valid in every lane
}

// ---------------------------------------------------------------------------
// Fused stencil + partial reduction.
//   pxy partial: sum over output pixels of |fx/Ec| + |fy/Ec|
//   pE  partial: sum over all 256x256 of pred_E
// One (pxy, pE) pair written per block -> deterministic, no atomics.
// ---------------------------------------------------------------------------
__global__ __launch_bounds__(256)
void inverse_loss_stencil(const float* __restrict__ pred_E,
                          const float* __restrict__ pred_v,
                          const float* __restrict__ data,
                          float* __restrict__ part_xy,
                          float* __restrict__ part_E) {
  const int nOut = RH_ * RW_;   // 64516
  const int nE   = H_ * W_;     // 65536
  const int step = gridDim.x * blockDim.x;

  float pxy = 0.0f, pE = 0.0f;

  for (int idx = blockIdx.x * blockDim.x + threadIdx.x; idx < nE; idx += step) {
    pE += pred_E[idx];
    if (idx < nOut) {
      const int i = idx / RW_;
      const int j = idx - i * RW_;
      float fx = 0.0f, fy = 0.0f, Ec = 0.0f;
#pragma unroll
      for (int di = 0; di < 3; ++di) {
#pragma unroll
        for (int dj = 0; dj < 3; ++dj) {
          const int r  = i + di;
          const int cc = j + dj;
          const float E = pred_E[r * W_ + cc];
          const float v = pred_v[r * W_ + cc];
          // displacement corners: data[(row*257+col)*2 + {0:ux,1:uy}]
          const float* d0 = data + 2 * (r * WP_ + cc);
          const float ux00 = d0[0], uy00 = d0[1];
          const float ux01 = d0[2], uy01 = d0[3];
          const float* d1 = data + 2 * ((r + 1) * WP_ + cc);
          const float ux10 = d1[0], uy10 = d1[1];
          const float ux11 = d1[2], uy11 = d1[3];

          const float exx = 50.0f * ((ux10 + ux11) - (ux00 + ux01));
          const float eyy = 50.0f * ((uy00 - uy01) + (uy10 - uy11));
          const float rxy = 50.0f * ((ux00 - ux01 + ux10 - ux11) +
                                     ((uy10 + uy11) - (uy00 + uy01)));

          const float frac = E / (1.0f - v * v);
          const float sxx  = (exx + v * eyy) * frac;
          const float syy  = (v * exx + eyy) * frac;
          const float sxy  = rxy * (1.0f - v) * 0.5f * frac;

          // wx_xx: row weight (-1,0,+1 by di) ; wx_xy: col weight (+1,0,-1 by dj)
          const float wr = (di == 0) ? -1.0f : ((di == 2) ? 1.0f : 0.0f);
          const float wc = (dj == 0) ?  1.0f : ((dj == 2) ? -1.0f : 0.0f);
          fx += wr * sxx + wc * sxy;
          fy += wc * syy + wr * sxy;
          Ec += E;
        }
      }
      pxy += fabsf(fx / Ec) + fabsf(fy / Ec);
    }
  }

  // wave-level reduction via WMMA (EXEC is all-1s here: uniform trip count,
  // divergence above is reconverged)
  const float wxy = wave_reduce_sum_wmma(pxy);
  const float wE  = wave_reduce_sum_wmma(pE);

  __shared__ float s_xy[8];
  __shared__ float s_E[8];
  const int lane = threadIdx.x & 31;
  const int wid  = threadIdx.x >> 5;
  if (lane == 0) { s_xy[wid] = wxy; s_E[wid] = wE; }
  __syncthreads();
  if (threadIdx.x == 0) {
    float bx = 0.0f, bE = 0.0f;
    const int nw = blockDim.x >> 5;
    for (int w = 0; w < nw; ++w) { bx += s_xy[w]; bE += s_E[w]; }
    part_xy[blockIdx.x] = bx;
    part_E[blockIdx.x]  = bE;
  }
}

// ---------------------------------------------------------------------------
// Finalize: one wave folds NB_ block partials (WMMA wave reduction again),
// assembles the scalar loss.
// ---------------------------------------------------------------------------
__global__ void inverse_loss_finalize(const float* __restrict__ part_xy,
                                      const float* __restrict__ part_E,
                                      float* __restrict__ out) {
  float sx = 0.0f, sE = 0.0f;
  for (int i = threadIdx.x; i < NB_; i += 32) {  // 128/32 = 4, uniform
    sx += part_xy[i];
    sE += part_E[i];
  }
  sx = wave_reduce_sum_wmma(sx);
  sE = wave_reduce_sum_wmma(sE);
  if (threadIdx.x == 0) {
    const float loss_xy = sx / (float)(RH_ * RW_);
    const float loss_e  = fabsf(sE / (float)(H_ * W_) - 0.25f);
    out[0] = loss_xy + loss_e * 0.01f;
  }
}

extern "C" void kernel_launch(void* const* d_in, const int* in_sizes, int n_in,
                              void* d_out, int out_size, void* d_ws, size_t ws_size,
                              hipStream_t stream) {
  (void)in_sizes; (void)n_in; (void)out_size; (void)ws_size;
  const float* pred_E = (const float*)d_in[0];
  const float* pred_v = (const float*)d_in[1];
  const float* data   = (const float*)d_in[2];
  float* out = (float*)d_out;

  float* part_xy = (float*)d_ws;        // NB_ floats
  float* part_E  = part_xy + NB_;       // NB_ floats (1 KB total scratch)

  inverse_loss_stencil<<<NB_, 256, 0, stream>>>(pred_E, pred_v, data,
                                                part_xy, part_E);
  inverse_loss_finalize<<<1, 32, 0, stream>>>(part_xy, part_E, out);
}